// BasicGCNSegmentation_48593259987372
// MI455X (gfx1250) — compile-verified
//
#include <hip/hip_runtime.h>
#include <hip/hip_bf16.h>

typedef __attribute__((ext_vector_type(2))) float v2f;
typedef __attribute__((ext_vector_type(8))) float v8f;

#define HDIM 128
#define GEMM_WAVES 8
#define XS_STRIDE 132   // 132 % 64 == 4 -> conflict-free A-frag reads across 16 lanes
#define WT_STRIDE 132   // even -> 8B-aligned b64 B-frag reads; halves land on disjoint banks
#define CAM_C 40

// ---------------------------------------------------------------- utilities
__global__ void zero_kernel(float* __restrict__ p, long n) {
  long i = (long)blockIdx.x * blockDim.x + threadIdx.x;
  const long stride = (long)gridDim.x * blockDim.x;
  for (; i < n; i += stride) p[i] = 0.0f;
}

__global__ void degree_kernel(const int* __restrict__ src, const int* __restrict__ dst,
                              float* __restrict__ outdeg, float* __restrict__ indeg, int E) {
  const int e = blockIdx.x * blockDim.x + threadIdx.x;
  if (e < E) {
    unsafeAtomicAdd(&outdeg[src[e]], 1.0f);   // global_atomic_add_f32
    unsafeAtomicAdd(&indeg[dst[e]], 1.0f);
  }
}

__global__ void norm_kernel(float* __restrict__ deg, int n) {
  const int i = blockIdx.x * blockDim.x + threadIdx.x;
  if (i < n) deg[i] = rsqrtf(fmaxf(deg[i], 1.0f));   // clip(deg,1)^-0.5
}

// ------------------------------------------------- GEMM: T = (X*out_norm) @ W
// One wave computes a 16(M) x 128(N) strip as 8 WMMA 16x16 tiles, K=128 in
// steps of 4 via V_WMMA_F32_16X16X4_F32 (exact f32, matches reference dtype).
// X tiles are staged with GLOBAL_LOAD_ASYNC_TO_LDS_B128 (ASYNCcnt path); the
// out_norm row scale is applied to the A-fragments in registers instead, so
// the staging is a pure DMA copy.
// A-frag layout per ISA 7.12.2: lanes 0-15 hold M=0..15 {K=k,k+1},
// lanes 16-31 hold M=0..15 {K=k+2,k+3}. B assumed mirrored (lane=N, half=K-pair).
__global__ __launch_bounds__(256) void gcn_gemm_wmma(
    const float* __restrict__ X, const float* __restrict__ W,
    const float* __restrict__ out_norm, float* __restrict__ T, int N) {
  __shared__ float smem[HDIM * WT_STRIDE + GEMM_WAVES * 16 * XS_STRIDE];
  float* Wt = smem;                                // W transposed: Wt[j][k]
  float* Xs = smem + HDIM * WT_STRIDE;             // per-wave raw X tile

  const int tid  = threadIdx.x;
  const int wave = tid >> 5;
  const int lane = tid & 31;
  const int half = lane >> 4;
  const int mr   = lane & 15;

  const int m0 = (blockIdx.x * GEMM_WAVES + wave) * 16;
  float* xs = &Xs[wave * 16 * XS_STRIDE];

  if (m0 < N) {
    // wave-relative LDS byte offset == low 32 bits of the flat LDS address
    // (ISA 10.2: LDS aperture truncates to addr[31:0])
    const unsigned ldsbase = (unsigned)(unsigned long long)(const void*)xs;
    #pragma unroll
    for (int r = 0; r < 16; ++r) {
      int row = m0 + r;
      if (row >= N) row = N - 1;                     // clamp: data unused (scale=0)
      const unsigned ldsoff = ldsbase + (unsigned)(r * XS_STRIDE * 4) + (unsigned)(lane * 16);
      const unsigned goff   = (unsigned)(row * HDIM + lane * 4) * 4u;
      // GVS mode: dsaddr from VGPR, global = SGPR base + 32-bit VGPR offset
      asm volatile("global_load_async_to_lds_b128 %0, %1, %2"
                   :: "v"(ldsoff), "v"(goff), "s"(X) : "memory");
    }
  }

  // stage W transposed (coalesced global reads) while the async DMA runs
  for (int idx = tid; idx < HDIM * HDIM; idx += 256) {
    const int k = idx >> 7, j = idx & (HDIM - 1);
    Wt[j * WT_STRIDE + k] = W[idx];
  }

  if (m0 < N) {
    asm volatile("s_wait_asynccnt 0x0" ::: "memory");   // X tile resident in LDS
  }
  __syncthreads();
  if (m0 >= N) return;             // wave-uniform; EXEC all-ones for WMMA below

  const int arow = m0 + mr;        // A-fragment row for this lane (both halves)
  const float rowscale = (arow < N) ? out_norm[arow] : 0.0f;

  v8f acc[8] = {};
  const float* xrow = xs + mr * XS_STRIDE;
  #pragma unroll 4
  for (int kk = 0; kk < HDIM; kk += 4) {
    const int k0 = kk + 2 * half;
    v2f a = *(const v2f*)(xrow + k0);                // ds_load_b64
    a.x *= rowscale;
    a.y *= rowscale;
    #pragma unroll
    for (int t = 0; t < 8; ++t) {
      const v2f b = *(const v2f*)(Wt + (t * 16 + mr) * WT_STRIDE + k0);
      acc[t] = __builtin_amdgcn_wmma_f32_16x16x4_f32(
          /*neg_a=*/false, a, /*neg_b=*/false, b,
          /*c_mod=*/(short)0, acc[t], /*reuse_a=*/false, /*reuse_b=*/false);
    }
  }

  // C/D layout: VGPR v, lanes 0-15 -> [M=v][N=lane]; lanes 16-31 -> [M=v+8]
  #pragma unroll
  for (int t = 0; t < 8; ++t) {
    #pragma unroll
    for (int v = 0; v < 8; ++v) {
      const int row = m0 + half * 8 + v;
      if (row < N) T[(size_t)row * HDIM + t * 16 + mr] = acc[t][v];
    }
  }
}

// ------------------------------------------- edge scatter: Y[dst] += T[src]
// one wave per edge; T/Y are L2-resident (51 MB each vs 192 MB L2)
__global__ __launch_bounds__(256) void scatter_kernel(
    const float* __restrict__ T, const int* __restrict__ src,
    const int* __restrict__ dst, float* __restrict__ Y, int E) {
  const int lane = threadIdx.x & 31;
  const int wave = threadIdx.x >> 5;
  for (int e = blockIdx.x * 8 + wave; e < E; e += gridDim.x * 8) {
    const int s = src[e];
    const int d = dst[e];
    const float4 v = *(const float4*)(T + (size_t)s * HDIM + lane * 4);
    float* y = Y + (size_t)d * HDIM + lane * 4;
    unsafeAtomicAdd(y + 0, v.x);
    unsafeAtomicAdd(y + 1, v.y);
    unsafeAtomicAdd(y + 2, v.z);
    unsafeAtomicAdd(y + 3, v.w);
  }
}

__global__ void relu_bias_kernel(const float* __restrict__ Y,
                                 const float* __restrict__ in_norm,
                                 const float* __restrict__ b,
                                 float* __restrict__ H, int total) {
  const int i = blockIdx.x * blockDim.x + threadIdx.x;
  if (i < total)
    H[i] = fmaxf(fmaf(Y[i], in_norm[i >> 7], b[i & (HDIM - 1)]), 0.0f);
}

// ------------------------------------------------------- readout + CAM
__global__ void mean_partial_kernel(const float* __restrict__ H,
                                    float* __restrict__ hgsum, int N) {
  const int j = threadIdx.x;   // 128 threads = 128 channels
  float s = 0.0f;
  for (int n = blockIdx.x; n < N; n += gridDim.x) s += H[(size_t)n * HDIM + j];
  unsafeAtomicAdd(&hgsum[j], s);
}

__global__ void seg_out_kernel(const float* __restrict__ hgsum,
                               const float* __restrict__ Wp,
                               const float* __restrict__ bp,
                               float* __restrict__ out, int N, int C) {
  const int c = threadIdx.x;
  if (c < C) {
    float s = 0.0f;
    for (int j = 0; j < HDIM; ++j) s += hgsum[j] * Wp[c * HDIM + j];
    out[c] = s / (float)N + bp[c];
  }
}

// cam[c*N+n] = dot(Wp[c,:], H[n,:]); Wp broadcast from LDS, stores coalesced in n
__global__ __launch_bounds__(256) void cam_kernel(const float* __restrict__ H,
                                                  const float* __restrict__ Wp,
                                                  float* __restrict__ cam, int N) {
  __shared__ float WpS[CAM_C * HDIM];
  for (int idx = threadIdx.x; idx < CAM_C * HDIM; idx += blockDim.x)
    WpS[idx] = Wp[idx];
  __syncthreads();
  const int n = blockIdx.x * blockDim.x + threadIdx.x;
  if (n >= N) return;

  float acc[CAM_C];
  #pragma unroll
  for (int c = 0; c < CAM_C; ++c) acc[c] = 0.0f;

  const float4* h4 = (const float4*)(H + (size_t)n * HDIM);
  #pragma unroll 1
  for (int ch = 0; ch < 4; ++ch) {              // 4 chunks of 32 channels
    float4 h[8];
    #pragma unroll
    for (int q = 0; q < 8; ++q) h[q] = h4[ch * 8 + q];
    #pragma unroll
    for (int c = 0; c < CAM_C; ++c) {
      const float4* w4 = (const float4*)(WpS + c * HDIM + ch * 32);
      float s = 0.0f;
      #pragma unroll
      for (int q = 0; q < 8; ++q) {
        const float4 w = w4[q];
        s += h[q].x * w.x + h[q].y * w.y + h[q].z * w.z + h[q].w * w.w;
      }
      acc[c] += s;
    }
  }
  #pragma unroll
  for (int c = 0; c < CAM_C; ++c) cam[(size_t)c * N + n] = acc[c];
}

// ---------------------------------------------------------------- launcher
static inline int zgrid(long n) {
  long g = (n + 255) / 256;
  return (int)(g > 4096 ? 4096 : g);
}

extern "C" void kernel_launch(void* const* d_in, const int* in_sizes, int n_in,
                              void* d_out, int out_size, void* d_ws, size_t ws_size,
                              hipStream_t stream) {
  const float* feats = (const float*)d_in[0];
  const float* W1 = (const float*)d_in[1];
  const float* b1 = (const float*)d_in[2];
  const float* W2 = (const float*)d_in[3];
  const float* b2 = (const float*)d_in[4];
  const float* W3 = (const float*)d_in[5];
  const float* b3 = (const float*)d_in[6];
  const float* Wp = (const float*)d_in[7];
  const float* bp = (const float*)d_in[8];
  const int*  src = (const int*)d_in[9];
  const int*  dst = (const int*)d_in[10];

  const int N = in_sizes[0] / HDIM;   // 100000
  const int E = in_sizes[9];          // 1600000
  const int C = in_sizes[8];          // 40

  // workspace layout (floats): out_norm | in_norm | T | Y | H | hgsum
  float* ws    = (float*)d_ws;
  float* outn  = ws;
  float* inn   = outn + N;
  float* T     = inn + N;
  float* Y     = T + (size_t)N * HDIM;
  float* H     = Y + (size_t)N * HDIM;
  float* hgsum = H + (size_t)N * HDIM;

  // degrees -> norms (outn/inn contiguous: zero & transform both at once)
  zero_kernel<<<zgrid(2L * N), 256, 0, stream>>>(outn, 2L * N);
  degree_kernel<<<(E + 255) / 256, 256, 0, stream>>>(src, dst, outn, inn, E);
  norm_kernel<<<(2 * N + 255) / 256, 256, 0, stream>>>(outn, 2 * N);

  const float* Wl[3] = {W1, W2, W3};
  const float* bl[3] = {b1, b2, b3};
  const long total = (long)N * HDIM;
  const int tiles = (N + 15) / 16;
  const int gemm_grid = (tiles + GEMM_WAVES - 1) / GEMM_WAVES;

  const float* Xin = feats;
  for (int l = 0; l < 3; ++l) {
    gcn_gemm_wmma<<<gemm_grid, 256, 0, stream>>>(Xin, Wl[l], outn, T, N);
    zero_kernel<<<zgrid(total), 256, 0, stream>>>(Y, total);
    scatter_kernel<<<4096, 256, 0, stream>>>(T, src, dst, Y, E);
    relu_bias_kernel<<<(int)((total + 255) / 256), 256, 0, stream>>>(Y, inn, bl[l], H, (int)total);
    Xin = H;
  }

  // readout
  zero_kernel<<<1, 128, 0, stream>>>(hgsum, HDIM);
  mean_partial_kernel<<<512, HDIM, 0, stream>>>(H, hgsum, N);
  seg_out_kernel<<<1, 64, 0, stream>>>(hgsum, Wp, bp, (float*)d_out, N, C);

  // CAM -> d_out[C .. C + C*N)
  cam_kernel<<<(N + 255) / 256, 256, 0, stream>>>(H, Wp, (float*)d_out + C, N);
}